// MultiHeadAttention_9259949491042
// MI455X (gfx1250) — compile-verified
//
#include <hip/hip_runtime.h>
#include <math.h>

typedef __attribute__((ext_vector_type(2))) float v2f;
typedef __attribute__((ext_vector_type(8))) float v8f;

#define BATCH  2
#define CIN    32
#define HH     32
#define WW     480
#define NHEADS 8
#define DH     4
#define QS     24
#define MF     8
#define WLEN   (QS + 2 * MF)    // 40
#define NB     (WW / QS)        // 20
#define NKEY   (HH * WLEN)      // 1280
#define NQ     (HH * QS)        // 768

// ---------------------------------------------------------------------------
// DPP16 row_xmask butterfly: xor-swap within each 16-lane row (wave32 has two
// rows, matching the two C-layout half-waves). Pure VALU, no LDS traffic.
// ---------------------------------------------------------------------------
template <int MASK>
__device__ __forceinline__ float dpp_xor(float x) {
    return __int_as_float(__builtin_amdgcn_update_dpp(
        0, __float_as_int(x), 0x160 | MASK, 0xF, 0xF, true));
}
__device__ __forceinline__ float rowmax16(float v) {
    v = fmaxf(v, dpp_xor<1>(v));
    v = fmaxf(v, dpp_xor<2>(v));
    v = fmaxf(v, dpp_xor<4>(v));
    v = fmaxf(v, dpp_xor<8>(v));
    return v;
}
__device__ __forceinline__ float rowsum16(float v) {
    v += dpp_xor<1>(v);
    v += dpp_xor<2>(v);
    v += dpp_xor<4>(v);
    v += dpp_xor<8>(v);
    return v;
}

// ---------------------------------------------------------------------------
// 3x3 conv, pad 1, as implicit GEMM on V_WMMA_F32_16X16X4_F32.
// M = COUT, K = CIN*9 = 288 (72 native K=4 steps), N = B*H*W = 30720.
// 4 waves / block, each wave owns a 16x16 output tile (shared M tile).
// ---------------------------------------------------------------------------
template <int COUT, bool HAS_BIAS>
__global__ __launch_bounds__(128) void conv3x3_wmma(
    const float* __restrict__ in,   // (B, CIN, H, W)
    const float* __restrict__ wgt,  // (COUT, CIN, 3, 3)
    const float* __restrict__ bias, // (COUT) or nullptr
    float* __restrict__ out)        // (B, COUT, H, W)
{
    const int lane = threadIdx.x & 31;
    const int wv   = threadIdx.x >> 5;
    const int half = lane >> 4;   // 0: lanes 0-15, 1: lanes 16-31
    const int l16  = lane & 15;

    const int mt = blockIdx.y * 16;               // out-channel tile base
    const int n0 = (blockIdx.x * 4 + wv) * 16;    // spatial tile base

    // This lane's B-matrix column (fixed across the K loop).
    const int ncol = n0 + l16;
    const int b    = ncol / (HH * WW);
    const int hw   = ncol % (HH * WW);
    const int h    = hw / WW;
    const int wc   = hw % WW;

    const int mrow = mt + l16;                     // this lane's A row

    // Incremental decode of k0 = k + 2*half and k1 = k0+1 into (ci, r=kh*3+kw).
    int ci0 = 0, r0 = half * 2;
    int ci1 = 0, r1 = half * 2 + 1;

    v8f acc = {};
    for (int k = 0; k < CIN * 9; k += 4) {
        const int k0 = k + half * 2;
        const int k1 = k0 + 1;

        // fp32 16x4 A layout: lanes 0-15 hold (K=k,k+1), lanes 16-31 (k+2,k+3)
        v2f a;
        a.x = wgt[mrow * (CIN * 9) + k0];
        a.y = wgt[mrow * (CIN * 9) + k1];

        v2f bv;
        {
            const int kh = r0 / 3, kw = r0 % 3;
            const int ih = h + kh - 1, iw = wc + kw - 1;
            bv.x = (ih >= 0 && ih < HH && iw >= 0 && iw < WW)
                       ? in[((b * CIN + ci0) * HH + ih) * WW + iw] : 0.f;
        }
        {
            const int kh = r1 / 3, kw = r1 % 3;
            const int ih = h + kh - 1, iw = wc + kw - 1;
            bv.y = (ih >= 0 && ih < HH && iw >= 0 && iw < WW)
                       ? in[((b * CIN + ci1) * HH + ih) * WW + iw] : 0.f;
        }
        acc = __builtin_amdgcn_wmma_f32_16x16x4_f32(false, a, false, bv,
                                                    (short)0, acc, false, false);
        r0 += 4; if (r0 >= 9) { r0 -= 9; ++ci0; }
        r1 += 4; if (r1 >= 9) { r1 -= 9; ++ci1; }
    }

    // C layout: VGPR j -> (M = mt + j + 8*half, N = ncol)
#pragma unroll
    for (int j = 0; j < 8; ++j) {
        const int m = mt + j + half * 8;
        float v = acc[j];
        if constexpr (HAS_BIAS) v += bias[m];
        out[((b * COUT + m) * HH + h) * WW + wc] = v;
    }
}

// ---------------------------------------------------------------------------
// Block-local attention: one workgroup per (batch, head, block) instance.
// logits = (0.5*q)(768x4) @ k(4x1280): one f32 WMMA per 16x16 tile (K=dh=4
// maps exactly onto the native 16x16x4 shape). Flash-style online softmax
// per wave (DPP row_xmask reductions), then P @ V via 4 more K=4 WMMAs.
// K/V window staged with GLOBAL_LOAD_ASYNC_TO_LDS (ASYNCcnt path).
// ---------------------------------------------------------------------------
__global__ __launch_bounds__(128) void attn_wmma(
    const float* __restrict__ qkv,  // (B, 96, H, W): per head 12 ch = q(4),k(4),v(4)
    float* __restrict__ outb)       // (B, 32, H, W)
{
    __shared__ float lds_k[DH][NKEY];   // 20 KB, K stored [d][n] for B-fragment gather
    __shared__ float lds_v[NKEY][DH];   // 20 KB, V stored [n][d]
    __shared__ float lds_p[4][256];     // per-wave P-tile staging (C->A layout)

    const int inst = blockIdx.x;
    const int blk  = inst % NB;
    const int head = (inst / NB) % NHEADS;
    const int b    = inst / (NB * NHEADS);

    const int lane = threadIdx.x & 31;
    const int wv   = threadIdx.x >> 5;
    const int half = lane >> 4;
    const int l16  = lane & 15;
    const int koff = half * 2;

    // ---- async staging of K (dh x NKEY) and V (NKEY x dh) into LDS ----
    // In-range columns: GLOBAL_LOAD_ASYNC_TO_LDS_B32 straight into LDS
    // (GVS mode: SGPR base + per-lane byte offset, per-lane LDS dest).
    // Out-of-range (padded) columns: plain zero DS stores (disjoint addrs).
    const int chbase = (b * 96 + head * 12) * HH * WW;
    for (int n = threadIdx.x; n < NKEY; n += 128) {
        const int h  = n / WLEN, wk = n % WLEN;
        const int w  = blk * QS + wk - MF;
        if (w >= 0 && w < WW) {
#pragma unroll
            for (int d = 0; d < DH; ++d) {
                const unsigned kgoff = (unsigned)((chbase + ((4 + d) * HH + h) * WW + w) * 4);
                const unsigned vgoff = (unsigned)((chbase + ((8 + d) * HH + h) * WW + w) * 4);
                const unsigned klds  = (unsigned)(uintptr_t)&lds_k[d][n];
                const unsigned vlds  = (unsigned)(uintptr_t)&lds_v[n][d];
                asm volatile("global_load_async_to_lds_b32 %0, %1, %2"
                             :: "v"(klds), "v"(kgoff), "s"(qkv) : "memory");
                asm volatile("global_load_async_to_lds_b32 %0, %1, %2"
                             :: "v"(vlds), "v"(vgoff), "s"(qkv) : "memory");
            }
        } else {
#pragma unroll
            for (int d = 0; d < DH; ++d) { lds_k[d][n] = 0.f; lds_v[n][d] = 0.f; }
        }
    }
    asm volatile("s_wait_asynccnt 0" ::: "memory");
    __syncthreads();

    // Each wave owns independent 16-row query tiles (48 tiles / 4 waves).
    for (int mtile = wv; mtile < NQ / 16; mtile += 4) {
        // A fragment (scaled q tile) -- constant over the key loop.
        const int Mg = mtile * 16 + l16;
        const int qh = Mg / QS, wq = Mg % QS;
        const int qw = blk * QS + wq;
        v2f aq;
        aq.x = 0.5f * qkv[((b * 96 + head * 12 + koff) * HH + qh) * WW + qw];
        aq.y = 0.5f * qkv[((b * 96 + head * 12 + koff + 1) * HH + qh) * WW + qw];

        v8f acc = {};
        float mrun[8], lrun[8];
#pragma unroll
        for (int j = 0; j < 8; ++j) { mrun[j] = -INFINITY; lrun[j] = 0.f; }

        int wk = l16;  // = (nt*16 + l16) % WLEN, maintained incrementally
        for (int nt = 0; nt < NKEY / 16; ++nt) {
            const int ncol = nt * 16 + l16;

            v2f bk;
            bk.x = lds_k[koff][ncol];
            bk.y = lds_k[koff + 1][ncol];

            v8f zero = {};
            v8f s = __builtin_amdgcn_wmma_f32_16x16x4_f32(false, aq, false, bk,
                                                          (short)0, zero, false, false);
            // reference's padded-column bias (only first/last blocks have pads)
            {
                const int wg = blk * QS + wk - MF;
                if (wg < 0 || wg >= WW) {
#pragma unroll
                    for (int j = 0; j < 8; ++j) s[j] = s[j] - 1e-9f;
                }
            }

            // Online softmax: row j of a C-layout tile lives across one
            // 16-lane DPP row, reduced with row_xmask butterflies.
            float p[8];
#pragma unroll
            for (int j = 0; j < 8; ++j) {
                const float mj    = rowmax16(s[j]);
                const float mnew  = fmaxf(mrun[j], mj);
                const float alpha = __expf(mrun[j] - mnew);  // first iter: exp(-inf)=0
                const float pj    = __expf(s[j] - mnew);
                const float lj    = rowsum16(pj);
                lrun[j] = lrun[j] * alpha + lj;
                mrun[j] = mnew;
                acc[j] *= alpha;
                p[j] = pj;
            }

            // Route P tile C-layout -> row-major LDS -> A-layout fragments.
            // LDS ops from one wave execute in order; just block compiler
            // reordering here.
#pragma unroll
            for (int j = 0; j < 8; ++j)
                lds_p[wv][(j + half * 8) * 16 + l16] = p[j];
            __asm__ volatile("" ::: "memory");

            // acc += P(16x16) @ Vpad(16x16): 4 native K=4 WMMAs.
#pragma unroll
            for (int kk = 0; kk < 16; kk += 4) {
                v2f ap, bv;
                ap.x = lds_p[wv][l16 * 16 + kk + koff];
                ap.y = lds_p[wv][l16 * 16 + kk + koff + 1];
                const int key = nt * 16 + kk + koff;
                bv.x = (l16 < DH) ? lds_v[key][l16] : 0.f;
                bv.y = (l16 < DH) ? lds_v[key + 1][l16] : 0.f;
                acc = __builtin_amdgcn_wmma_f32_16x16x4_f32(false, ap, false, bv,
                                                            (short)0, acc, false, false);
            }
            __asm__ volatile("" ::: "memory");

            wk += 16; if (wk >= WLEN) wk -= WLEN;
        }

        // Columns 0..3 of the accumulator hold the dh outputs.
        if (l16 < DH) {
#pragma unroll
            for (int j = 0; j < 8; ++j) {
                const int M = mtile * 16 + j + half * 8;
                const int h = M / QS, wq2 = M % QS;
                const int w = blk * QS + wq2;
                const int c = head * DH + l16;
                outb[((b * (NHEADS * DH) + c) * HH + h) * WW + w] = acc[j] / lrun[j];
            }
        }
    }
}

// ---------------------------------------------------------------------------
extern "C" void kernel_launch(void* const* d_in, const int* in_sizes, int n_in,
                              void* d_out, int out_size, void* d_ws, size_t ws_size,
                              hipStream_t stream)
{
    const float* x     = (const float*)d_in[0];  // (2,32,32,480)
    const float* qkv_w = (const float*)d_in[1];  // (96,32,3,3)
    const float* qkv_b = (const float*)d_in[2];  // (96)
    const float* out_w = (const float*)d_in[3];  // (32,32,3,3)
    float*       outp  = (float*)d_out;          // (2,32,32,480)

    // workspace: qkv buffer (11.8 MB) + attention output (3.9 MB)
    float* qkvbuf  = (float*)d_ws;
    float* attnbuf = qkvbuf + (size_t)BATCH * 96 * HH * WW;

    const dim3 blk(128);
    const int  ntiles = (BATCH * HH * WW) / 64;  // 480 spatial tile groups

    conv3x3_wmma<96, true><<<dim3(ntiles, 96 / 16), blk, 0, stream>>>(
        x, qkv_w, qkv_b, qkvbuf);

    attn_wmma<<<dim3(BATCH * NHEADS * NB), blk, 0, stream>>>(qkvbuf, attnbuf);

    conv3x3_wmma<32, false><<<dim3(ntiles, 32 / 16), blk, 0, stream>>>(
        attnbuf, out_w, nullptr, outp);
}